// MambaVisionMultiHead_33861522162256
// MI455X (gfx1250) — compile-verified
//
#include <hip/hip_runtime.h>
#include <hip/hip_bf16.h>

typedef __attribute__((ext_vector_type(2))) float v2f;
typedef __attribute__((ext_vector_type(8))) float v8f;

// ---------------------------------------------------------------------------
// K1: fused conv2d + (lin_w row 31) contraction -> u31 (512 x 128)
// t[b,pq] = sum_ij lin_w[31,ij] * x_pad[b, i+p-1, j+q-1]
// u31[b,e] = sum_pq conv_w[e,pq]*t[b,pq] + conv_b[e]*sum(lin_w[31]) + lin_b[31]
// one block per image b, 128 threads (one per e after reduction)
// ---------------------------------------------------------------------------
__global__ void k1_front(const float* __restrict__ x,
                         const float* __restrict__ conv_w,
                         const float* __restrict__ conv_b,
                         const float* __restrict__ lin_w,
                         const float* __restrict__ lin_b,
                         float* __restrict__ u31) {
  int b = blockIdx.x;
  int t = threadIdx.x;                      // 0..127
  const float* xb  = x + b * 1024;
  const float* w31 = lin_w + 31 * 1024;
  float acc[9];
#pragma unroll
  for (int i = 0; i < 9; ++i) acc[i] = 0.f;
  float wsum = 0.f;
  for (int k = 0; k < 8; ++k) {
    int ij = t + 128 * k;                   // 0..1023
    int i = ij >> 5, j = ij & 31;
    float w = w31[ij];
    wsum += w;
#pragma unroll
    for (int p = 0; p < 3; ++p) {
      int ii = i + p - 1;
      if (ii < 0 || ii > 31) continue;
#pragma unroll
      for (int q = 0; q < 3; ++q) {
        int jj = j + q - 1;
        if (jj < 0 || jj > 31) continue;
        acc[p * 3 + q] += w * xb[ii * 32 + jj];
      }
    }
  }
  __shared__ float red[10][128];
#pragma unroll
  for (int i = 0; i < 9; ++i) red[i][t] = acc[i];
  red[9][t] = wsum;
  __syncthreads();
  for (int stride = 64; stride > 0; stride >>= 1) {
    if (t < stride) {
#pragma unroll
      for (int i = 0; i < 10; ++i) red[i][t] += red[i][t + stride];
    }
    __syncthreads();
  }
  float s31 = red[9][0];
  float a2 = conv_b[t] * s31 + lin_b[31];
#pragma unroll
  for (int pq = 0; pq < 9; ++pq) a2 += conv_w[t * 9 + pq] * red[pq][0];
  u31[b * 128 + t] = a2;
}

// ---------------------------------------------------------------------------
// f32 GEMM via V_WMMA_F32_16X16X4_F32:  Out[M,N] = act(X[M,K] @ W[N,K]^T + bias)
// One wave (32 threads) per 16x16 output tile; K-loop of 4.
// A frag layout: lane L -> M = L%16, K = v + 2*(L/16)   (v = 0..1)
// B frag layout: lane L -> N = L%16, K = v + 2*(L/16)   (B = W^T, row-major W)
// D layout: vgpr r -> M = r + 8*(L/16), N = L%16
// ACT: 0 = none, 1 = softplus
// ---------------------------------------------------------------------------
template <int ACT>
__global__ void gemm_xwT_wmma(const float* __restrict__ X,
                              const float* __restrict__ W,
                              const float* __restrict__ bias,
                              float* __restrict__ Out,
                              int M, int N, int K) {
  int ntn  = N >> 4;
  int tile = blockIdx.x;
  int tm = tile / ntn, tn = tile % ntn;
  int lane = threadIdx.x;                   // 0..31, wave32
  int hh   = lane >> 4;                     // 0 or 1
  int l16  = lane & 15;
  const float* Xrow = X + (tm * 16 + l16) * K + 2 * hh;
  const float* Wrow = W + (tn * 16 + l16) * K + 2 * hh;
  v8f c = {0.f, 0.f, 0.f, 0.f, 0.f, 0.f, 0.f, 0.f};
  for (int kk = 0; kk < K; kk += 4) {
    v2f a, bfr;
    a[0]   = Xrow[kk];
    a[1]   = Xrow[kk + 1];
    bfr[0] = Wrow[kk];
    bfr[1] = Wrow[kk + 1];
    c = __builtin_amdgcn_wmma_f32_16x16x4_f32(
        /*neg_a=*/false, a, /*neg_b=*/false, bfr,
        /*c_mod=*/(short)0, c, /*reuse_a=*/false, /*reuse_b=*/false);
  }
  int n = tn * 16 + l16;
  float bv = bias ? bias[n] : 0.f;
#pragma unroll
  for (int r = 0; r < 8; ++r) {
    int m = tm * 16 + r + 8 * hh;
    float v = c[r] + bv;
    if (ACT == 1) v = (v > 20.f) ? v : log1pf(__expf(v));  // softplus
    Out[m * N + n] = v;
  }
}

// ---------------------------------------------------------------------------
// K3: depthwise causal conv1d (K=4) + SiLU.  xm[l,c] = xz[l*512 + c]
// ---------------------------------------------------------------------------
__global__ void k3_conv1d(const float* __restrict__ xz,
                          const float* __restrict__ w1,
                          const float* __restrict__ b1,
                          float* __restrict__ xs) {
  int idx = blockIdx.x * blockDim.x + threadIdx.x;
  if (idx >= 512 * 256) return;
  int l = idx >> 8, cch = idx & 255;
  float acc = b1[cch];
#pragma unroll
  for (int k = 0; k < 4; ++k) {
    int ll = l - 3 + k;
    if (ll >= 0) acc += xz[ll * 512 + cch] * w1[cch * 4 + k];
  }
  xs[idx] = acc / (1.f + __expf(-acc));     // silu
}

// Mdt[d,c] = sum_{r<8} W_dt[d,r] * W_x[r,c]   (fold dt_rank bottleneck)
__global__ void k_mdt(const float* __restrict__ W_dt,
                      const float* __restrict__ W_x,
                      float* __restrict__ Mdt) {
  int idx = blockIdx.x * blockDim.x + threadIdx.x;
  if (idx >= 256 * 256) return;
  int d = idx >> 8, cch = idx & 255;
  float acc = 0.f;
#pragma unroll
  for (int r = 0; r < 8; ++r) acc += W_dt[d * 8 + r] * W_x[r * 256 + cch];
  Mdt[idx] = acc;
}

// Acoef = -exp(A_log)
__global__ void k_acoef(const float* __restrict__ A_log, float* __restrict__ Ac) {
  int idx = blockIdx.x * blockDim.x + threadIdx.x;
  if (idx < 256 * 256) Ac[idx] = -__expf(A_log[idx]);
}

// ---------------------------------------------------------------------------
// K5: selective scan. Single workgroup, 1024 threads, h (256x256 f32) in LDS.
// thread tid: d = tid>>2, owns s-range [(tid&3)*64, +64)
// h[d,s] = exp(delta[l,d]*A[d,s])*h + delta[l,d]*xs[l,d]*B[l,s]
// ys[l,d] = sum_s h[d,s]*C[l,s]
// ---------------------------------------------------------------------------
__global__ void k5_scan(const float* __restrict__ delta,
                        const float* __restrict__ xs,
                        const float* __restrict__ BC,
                        const float* __restrict__ Ac,
                        float* __restrict__ ysum) {
  extern __shared__ float smem[];
  float* h    = smem;               // 65536 floats = 256 KB
  float* Brow = smem + 65536;       // 256
  float* Crow = Brow + 256;         // 256
  float* dlv  = Crow + 256;         // 256
  float* dxv  = dlv + 256;          // 256
  int tid = threadIdx.x;            // 0..1023
  int d   = tid >> 2;
  int sg  = tid & 3;
  int sb  = sg * 64;
  float* hp = h + d * 256 + sb;
  const float* Ap = Ac + d * 256 + sb;
#pragma unroll 8
  for (int i = 0; i < 64; ++i) hp[i] = 0.f;
  __syncthreads();
  for (int l = 0; l < 512; ++l) {
    if (tid < 512) {
      float v = BC[l * 512 + tid];
      if (tid < 256) Brow[tid] = v; else Crow[tid - 256] = v;
    } else if (tid < 768) {
      int dd = tid - 512;
      float dv = delta[l * 256 + dd];
      dlv[dd] = dv;
      dxv[dd] = dv * xs[l * 256 + dd];
    }
    __syncthreads();
    float dl = dlv[d];
    float dx = dxv[d];
    float part = 0.f;
#pragma unroll 4
    for (int i = 0; i < 64; ++i) {
      int s = sb + i;
      float hv = hp[i];
      hv = __expf(dl * Ap[i]) * hv + dx * Brow[s];
      hp[i] = hv;
      part += hv * Crow[s];
    }
    part += __shfl_xor(part, 1);
    part += __shfl_xor(part, 2);
    if (sg == 0) ysum[l * 256 + d] = part;
    __syncthreads();
  }
}

// K6a: y = (ys + xs*Dp) * silu(zg);  zg[l,d] = xz[l*512 + 256 + d]
__global__ void k6_gate(const float* __restrict__ ysum,
                        const float* __restrict__ xs,
                        const float* __restrict__ xz,
                        const float* __restrict__ Dp,
                        float* __restrict__ y) {
  int idx = blockIdx.x * blockDim.x + threadIdx.x;
  if (idx >= 512 * 256) return;
  int l = idx >> 8, d = idx & 255;
  float zg = xz[l * 512 + 256 + d];
  float sil = zg / (1.f + __expf(-zg));
  y[idx] = (ysum[idx] + xs[idx] * Dp[d]) * sil;
}

// K7: three head GEMVs, outputs concatenated: (512x1),(512x3),(512x4)
__global__ void k7_heads(const float* __restrict__ feat,
                         const float* __restrict__ We, const float* __restrict__ be,
                         const float* __restrict__ Wa, const float* __restrict__ ba,
                         const float* __restrict__ Wq, const float* __restrict__ bq,
                         float* __restrict__ out) {
  int idx = blockIdx.x * blockDim.x + threadIdx.x;
  if (idx >= 512 * 8) return;
  int b = idx >> 3, j = idx & 7;
  const float* f = feat + b * 128;
  const float* w;
  float bias;
  float* dst;
  if (j == 0)      { w = We;               bias = be[0];     dst = out + b; }
  else if (j < 4)  { w = Wa + (j - 1) * 128; bias = ba[j - 1]; dst = out + 512 + b * 3 + (j - 1); }
  else             { w = Wq + (j - 4) * 128; bias = bq[j - 4]; dst = out + 512 + 1536 + b * 4 + (j - 4); }
  float acc = bias;
#pragma unroll 8
  for (int k = 0; k < 128; ++k) acc += f[k] * w[k];
  *dst = acc;
}

extern "C" void kernel_launch(void* const* d_in, const int* in_sizes, int n_in,
                              void* d_out, int out_size, void* d_ws, size_t ws_size,
                              hipStream_t stream) {
  const float* x       = (const float*)d_in[0];
  const float* conv_w  = (const float*)d_in[1];
  const float* conv_b  = (const float*)d_in[2];
  const float* lin_w   = (const float*)d_in[3];
  const float* lin_b   = (const float*)d_in[4];
  const float* W_in    = (const float*)d_in[5];
  const float* conv1dw = (const float*)d_in[6];
  const float* conv1db = (const float*)d_in[7];
  const float* W_x     = (const float*)d_in[8];
  const float* W_dt    = (const float*)d_in[9];
  const float* b_dt    = (const float*)d_in[10];
  const float* A_log   = (const float*)d_in[11];
  const float* Dp      = (const float*)d_in[12];
  const float* W_out   = (const float*)d_in[13];
  const float* We      = (const float*)d_in[14];
  const float* be      = (const float*)d_in[15];
  const float* Wa      = (const float*)d_in[16];
  const float* ba      = (const float*)d_in[17];
  const float* Wq      = (const float*)d_in[18];
  const float* bq      = (const float*)d_in[19];
  float* out = (float*)d_out;

  float* ws = (float*)d_ws;
  float* u31   = ws;                 // 512*128
  float* xz    = u31   + 65536;      // 512*512
  float* xs    = xz    + 262144;     // 512*256
  float* Mdt   = xs    + 131072;     // 256*256
  float* delta = Mdt   + 65536;      // 512*256
  float* BC    = delta + 131072;     // 512*512
  float* Ac    = BC    + 262144;     // 256*256
  float* ysum  = Ac    + 65536;      // 512*256
  float* y     = ysum  + 131072;     // 512*256
  float* feat  = y     + 131072;     // 512*128

  // 1) conv2d + lin(row 31) -> u31 (512x128)
  k1_front<<<512, 128, 0, stream>>>(x, conv_w, conv_b, lin_w, lin_b, u31);

  // 2) xz = u31 @ W_in^T  (512x512, K=128)  [WMMA]
  gemm_xwT_wmma<0><<<(512 / 16) * (512 / 16), 32, 0, stream>>>(
      u31, W_in, nullptr, xz, 512, 512, 128);

  // 3) depthwise causal conv1d + silu -> xs (512x256)
  k3_conv1d<<<512, 256, 0, stream>>>(xz, conv1dw, conv1db, xs);

  // 4a) Mdt = W_dt @ W_x[:8]  (256x256)
  k_mdt<<<256, 256, 0, stream>>>(W_dt, W_x, Mdt);

  // 4b) delta = softplus(xs @ Mdt^T + b_dt)  (512x256, K=256)  [WMMA]
  gemm_xwT_wmma<1><<<(512 / 16) * (256 / 16), 32, 0, stream>>>(
      xs, Mdt, b_dt, delta, 512, 256, 256);

  // 4c) BC = xs @ W_x[8:]^T  (512x512, K=256)  [WMMA]
  gemm_xwT_wmma<0><<<(512 / 16) * (512 / 16), 32, 0, stream>>>(
      xs, W_x + 8 * 256, nullptr, BC, 512, 512, 256);

  // 5) A coefficients, then the sequential selective scan (h in 256KB LDS)
  k_acoef<<<256, 256, 0, stream>>>(A_log, Ac);
  size_t smem = (size_t)(65536 + 4 * 256) * sizeof(float);   // ~260 KB of 320 KB
  k5_scan<<<1, 1024, smem, stream>>>(delta, xs, BC, Ac, ysum);

  // 6) gate + output projection: feat = y @ W_out^T (512x128, K=256) [WMMA]
  k6_gate<<<512, 256, 0, stream>>>(ysum, xs, xz, Dp, y);
  gemm_xwT_wmma<0><<<(512 / 16) * (128 / 16), 32, 0, stream>>>(
      y, W_out, nullptr, feat, 512, 128, 256);

  // 7) heads -> d_out (concat 512x1, 512x3, 512x4)
  k7_heads<<<16, 256, 0, stream>>>(feat, We, be, Wa, ba, Wq, bq, out);
}